// DynamicSparseAttention_74577812127897
// MI455X (gfx1250) — compile-verified
//
#include <hip/hip_runtime.h>
#include <math.h>

typedef float v2f __attribute__((ext_vector_type(2)));
typedef float v8f __attribute__((ext_vector_type(8)));

#define B_SZ 4
#define T_SEQ 1024
#define C_DIM 1024
#define HEADS 16
#define HD 64
#define KSLOT 16

#define LDS_STRIDE 20   // 64 rows x 20 floats: conflict-free b64 frag reads, 16B-aligned b128

// per-lane async copy of 16B global -> LDS (ASYNCcnt-tracked, no VGPR staging)
__device__ __forceinline__ void async_b128(unsigned lds_off, const float* gptr) {
    asm volatile("global_load_async_to_lds_b128 %0, %1, off"
                 :: "v"(lds_off), "v"(gptr) : "memory");
}
__device__ __forceinline__ void async_b128_o16(unsigned lds_off, const float* gptr) {
    // INST_OFFSET applies to both LDS and global address -> copies the next 16B
    asm volatile("global_load_async_to_lds_b128 %0, %1, off offset:16"
                 :: "v"(lds_off), "v"(gptr) : "memory");
}

// ---------------------------------------------------------------------------
// Big GEMM: C[M,N] = A[M,K] @ W[N,K]^T, M%64==0, N%64==0, K%16==0.
// 128 threads = 4 waves; block computes a 64x64 macro-tile.
// Double-buffered 64x16 K-chunks copied with GLOBAL_LOAD_ASYNC_TO_LDS_B128,
// overlapped with the WMMA consumption of the other buffer.
// ---------------------------------------------------------------------------
__global__ __launch_bounds__(128) void gemm_xwt_f32_wmma_t64(
    const float* __restrict__ A, const float* __restrict__ W,
    float* __restrict__ C, int K, int lda, int ldw, int ldc)
{
    const int n0   = blockIdx.x * 64;
    const int m0   = blockIdx.y * 64;
    const int tid  = threadIdx.x;
    const int wave = tid >> 5;          // 0..3 -> N subtile
    const int lane = tid & 31;
    const int l    = lane & 15;
    const int half = lane >> 4;         // lanes 0-15: K pair {0,1}; 16-31: {2,3}

    __shared__ float sA[2][64 * LDS_STRIDE];
    __shared__ float sB[2][64 * LDS_STRIDE];

    // staging map: 128 threads x (half-row of 8 floats = 2 x b128) covers 64x16
    const int srow = tid >> 1;          // 0..63
    const int scol = (tid & 1) * 8;     // 0 or 8

    const float* gA = A + (size_t)(m0 + srow) * lda + scol;
    const float* gW = W + (size_t)(n0 + srow) * ldw + scol;

    const unsigned aoff0 = (unsigned)(uintptr_t)&sA[0][srow * LDS_STRIDE + scol];
    const unsigned aoff1 = (unsigned)(uintptr_t)&sA[1][srow * LDS_STRIDE + scol];
    const unsigned boff0 = (unsigned)(uintptr_t)&sB[0][srow * LDS_STRIDE + scol];
    const unsigned boff1 = (unsigned)(uintptr_t)&sB[1][srow * LDS_STRIDE + scol];

    v8f acc[4] = {v8f{}, v8f{}, v8f{}, v8f{}};

    // prologue: async-copy chunk 0 into buffer 0
    async_b128(aoff0, gA);      async_b128_o16(aoff0, gA);
    async_b128(boff0, gW);      async_b128_o16(boff0, gW);

    for (int k0 = 0; k0 < K; k0 += 16) {
        const int cur = (k0 >> 4) & 1;
        if (k0 + 16 < K) {
            // issue next chunk into the other buffer, then retire current chunk
            const float* pa = gA + k0 + 16;
            const float* pb = gW + k0 + 16;
            const unsigned na = cur ? aoff0 : aoff1;
            const unsigned nb = cur ? boff0 : boff1;
            async_b128(na, pa);  async_b128_o16(na, pa);
            async_b128(nb, pb);  async_b128_o16(nb, pb);
            asm volatile("s_wait_asynccnt 4" ::: "memory");
        } else {
            asm volatile("s_wait_asynccnt 0" ::: "memory");
        }
        __syncthreads();   // all waves' chunk data now resident in LDS

        const float* cA = sA[cur];
        const float* cB = sB[cur];
        #pragma unroll
        for (int kk = 0; kk < 16; kk += 4) {
            v2f bf = *(const v2f*)(cB + (wave * 16 + l) * LDS_STRIDE + kk + 2 * half);
            #pragma unroll
            for (int ms = 0; ms < 4; ++ms) {
                v2f af = *(const v2f*)(cA + (ms * 16 + l) * LDS_STRIDE + kk + 2 * half);
                acc[ms] = __builtin_amdgcn_wmma_f32_16x16x4_f32(
                    false, af, false, bf, (short)0, acc[ms], false, false);
            }
        }
        __syncthreads();   // buffer free for reuse two chunks later
    }

    // store: lane l -> column n0 + wave*16 + l; VGPR i -> row + 8*half + i
    const int ccol = n0 + wave * 16 + l;
    #pragma unroll
    for (int ms = 0; ms < 4; ++ms) {
        float* crow = C + (size_t)(m0 + ms * 16 + 8 * half) * ldc + ccol;
        #pragma unroll
        for (int i = 0; i < 8; ++i)
            crow[(size_t)i * ldc] = acc[ms][i];
    }
}

// ---------------------------------------------------------------------------
// Small GEMM (one wave per 16x16 tile) for the 16-row KV projections.
// ---------------------------------------------------------------------------
__global__ __launch_bounds__(32) void gemm_xwt_f32_wmma(
    const float* __restrict__ A, const float* __restrict__ W,
    float* __restrict__ C, int K, int lda, int ldw, int ldc)
{
    const int n0   = blockIdx.x * 16;
    const int m0   = blockIdx.y * 16;
    const int lane = threadIdx.x & 31;
    const int l    = lane & 15;
    const int half = lane >> 4;

    const float* arow = A + (size_t)(m0 + l) * lda + 2 * half;
    const float* wrow = W + (size_t)(n0 + l) * ldw + 2 * half;

    v8f acc0 = {};
    v8f acc1 = {};
    for (int k0 = 0; k0 < K; k0 += 8) {
        v2f a0 = *(const v2f*)(arow + k0);
        v2f b0 = *(const v2f*)(wrow + k0);
        v2f a1 = *(const v2f*)(arow + k0 + 4);
        v2f b1 = *(const v2f*)(wrow + k0 + 4);
        acc0 = __builtin_amdgcn_wmma_f32_16x16x4_f32(
            false, a0, false, b0, (short)0, acc0, false, false);
        acc1 = __builtin_amdgcn_wmma_f32_16x16x4_f32(
            false, a1, false, b1, (short)0, acc1, false, false);
    }
    v8f acc = acc0 + acc1;

    float* crow = C + (size_t)(m0 + 8 * half) * ldc + (n0 + l);
    #pragma unroll
    for (int i = 0; i < 8; ++i)
        crow[(size_t)i * ldc] = acc[i];
}

// ---------------------------------------------------------------------------
// Attention epilogue. One block per (b,t): 256 threads = 16 heads x 16 slots.
// KV layout: [b][slot 0..15][2C], cols 0..C-1 = K-proj, C..2C-1 = V-proj.
// Top-k indices are provably arange(16); only mask j<=t + softmax remain.
// ---------------------------------------------------------------------------
__global__ __launch_bounds__(256) void attn_epilogue(
    const float* __restrict__ Q, const float* __restrict__ KV,
    float* __restrict__ O)
{
    const int bt  = blockIdx.x;
    const int b   = bt >> 10;           // T = 1024
    const int t   = bt & (T_SEQ - 1);
    const int tid = threadIdx.x;
    const int h   = tid >> 4;
    const int j   = tid & 15;

    __shared__ float sw[HEADS * KSLOT];

    const float* q  = Q  + (size_t)bt * C_DIM + h * HD;
    const float* kj = KV + ((size_t)(b * KSLOT + j)) * (2 * C_DIM) + h * HD;
    float dot = 0.f;
    #pragma unroll
    for (int d = 0; d < HD; d += 4) {
        float4 qa = *(const float4*)(q + d);
        float4 kb = *(const float4*)(kj + d);
        dot += qa.x * kb.x + qa.y * kb.y + qa.z * kb.z + qa.w * kb.w;
    }
    float logit = dot * 0.125f;          // 1/sqrt(64)
    if (j > t) logit = -INFINITY;
    sw[tid] = logit;
    __syncthreads();

    float mx = -INFINITY;
    #pragma unroll
    for (int jj = 0; jj < KSLOT; ++jj) mx = fmaxf(mx, sw[h * 16 + jj]);
    float sum = 0.f;
    #pragma unroll
    for (int jj = 0; jj < KSLOT; ++jj) sum += expf(sw[h * 16 + jj] - mx);
    float w = expf(logit - mx) / sum;
    __syncthreads();
    sw[tid] = w;
    __syncthreads();

    const int c0 = tid * 4;
    const int hc = c0 >> 6;
    const float* vbase = KV + (size_t)b * KSLOT * 2 * C_DIM + C_DIM + c0;
    float4 accv = make_float4(0.f, 0.f, 0.f, 0.f);
    #pragma unroll
    for (int jj = 0; jj < KSLOT; ++jj) {
        float wj = sw[hc * 16 + jj];
        float4 v = *(const float4*)(vbase + (size_t)jj * 2 * C_DIM);
        accv.x += wj * v.x; accv.y += wj * v.y;
        accv.z += wj * v.z; accv.w += wj * v.w;
    }
    *(float4*)(O + (size_t)bt * C_DIM + c0) = accv;
}

extern "C" void kernel_launch(void* const* d_in, const int* in_sizes, int n_in,
                              void* d_out, int out_size, void* d_ws, size_t ws_size,
                              hipStream_t stream) {
    (void)in_sizes; (void)n_in; (void)out_size; (void)ws_size;

    const float* x      = (const float*)d_in[0];   // (B,T,C)
    const float* W_attn = (const float*)d_in[1];   // (3C,C)
    const float* W_proj = (const float*)d_in[2];   // (C,C)
    // d_in[3] = W_rel: unused (top-k indices are data-independent)

    float* out = (float*)d_out;

    float* Q  = (float*)d_ws;                                 // B*T*C
    float* KV = Q  + (size_t)B_SZ * T_SEQ * C_DIM;            // B*16*2C
    float* O  = KV + (size_t)B_SZ * KSLOT * 2 * C_DIM;        // B*T*C

    // 1) Q = x @ Wq^T  (64x64 macro-tiles, async double-buffered LDS)
    gemm_xwt_f32_wmma_t64<<<dim3(C_DIM / 64, (B_SZ * T_SEQ) / 64), 128, 0, stream>>>(
        x, W_attn, Q, C_DIM, C_DIM, C_DIM, C_DIM);

    // 2) per batch: KV[b] (16 x 2C) = x[b,0:16,:] @ W_kv^T
    for (int b = 0; b < B_SZ; ++b) {
        gemm_xwt_f32_wmma<<<dim3((2 * C_DIM) / 16, 1), 32, 0, stream>>>(
            x + (size_t)b * T_SEQ * C_DIM,
            W_attn + (size_t)C_DIM * C_DIM,
            KV + (size_t)b * KSLOT * 2 * C_DIM,
            C_DIM, C_DIM, C_DIM, 2 * C_DIM);
    }

    // 3) masked-softmax attention over the fixed 16 slots
    attn_epilogue<<<B_SZ * T_SEQ, 256, 0, stream>>>(Q, KV, O);

    // 4) out = O @ W_proj^T
    gemm_xwt_f32_wmma_t64<<<dim3(C_DIM / 64, (B_SZ * T_SEQ) / 64), 128, 0, stream>>>(
        O, W_proj, out, C_DIM, C_DIM, C_DIM, C_DIM);
}